// InteractionNet_4861902979305
// MI455X (gfx1250) — compile-verified
//
#include <hip/hip_runtime.h>

typedef float v2f __attribute__((ext_vector_type(2)));
typedef float v8f __attribute__((ext_vector_type(8)));

#define FDIM 128
#define N1D  256
#define N2D  256
#define SP   132      // padded LDS row stride for F-wide arrays (kills stride-128 bank conflicts)
#define SMP  260      // padded LDS row stride for mask rows
#define ROWS 16       // i-rows per workgroup

#define OFF_MX2  0                     // 256 x SP  (Mx2[b], 135168 B)
#define OFF_X1   (N2D * SP)            // 16 x SP   (x1 tile / GRU hidden)
#define OFF_X    (OFF_X1 + ROWS * SP)  // 16 x SP   (m2, then x = relu(m1+m2))
#define OFF_MASK (OFF_X + ROWS * SP)   // 16 x SMP  (valid_edge tile)
#define SMEM_FLOATS (OFF_MASK + ROWS * SMP)   // 42176 floats = 168704 bytes < 320 KB

// D = A(16x4 f32) * B(4x16 f32) + C(16x16 f32), exact fp32 WMMA
__device__ __forceinline__ v8f wmma_k4(v2f a, v2f b, v8f c) {
#if defined(__gfx1250__)
  // 8 args: (neg_a, A, neg_b, B, c_mod, C, reuse_a, reuse_b)
  return __builtin_amdgcn_wmma_f32_16x16x4_f32(false, a, false, b, (short)0, c, false, false);
#else
  (void)a; (void)b;
  return c;
#endif
}

// Async memory->LDS B128 copy (ASYNCcnt-tracked, no VGPR round trip).
// lds_byte_off: byte offset into the workgroup's LDS (dynamic LDS starts at 0
// here since the kernel declares no static __shared__).
__device__ __forceinline__ void stage_b128(float* smem_base, unsigned lds_byte_off,
                                           const float* g) {
#if defined(__gfx1250__)
  (void)smem_base;
  asm volatile("global_load_async_to_lds_b128 %0, %1, off"
               :: "v"(lds_byte_off), "v"(g)
               : "memory");
#else
  float4* d = (float4*)((char*)smem_base + lds_byte_off);
  *d = *(const float4*)g;
#endif
}

__device__ __forceinline__ void wait_async0() {
#if defined(__gfx1250__)
  asm volatile("s_wait_asynccnt 0x0" ::: "memory");
#endif
}

__device__ __forceinline__ bool wave_any_nonzero(float m) {
#if defined(__gfx1250__)
  return __builtin_amdgcn_ballot_w32(m != 0.0f) != 0u;
#else
  return m != 0.0f;
#endif
}

__device__ __forceinline__ float sigm(float x) { return 1.0f / (1.0f + __expf(-x)); }
__device__ __forceinline__ float tanh_fast(float x) {
  x = fminf(15.0f, fmaxf(-15.0f, x));
  const float e = __expf(-2.0f * x);
  return (1.0f - e) / (1.0f + e);
}

__global__ __launch_bounds__(256) void inet_fused(
    const float* __restrict__ x1,  const float* __restrict__ x2,
    const float* __restrict__ ve,
    const float* __restrict__ Ww,  const float* __restrict__ Wb,
    const float* __restrict__ Mw,  const float* __restrict__ Mb,
    const float* __restrict__ wih, const float* __restrict__ whh,
    const float* __restrict__ bih, const float* __restrict__ bhh,
    float* __restrict__ out)
{
  extern __shared__ float smem[];
  float* sMx2 = smem + OFF_MX2;
  float* sX1  = smem + OFF_X1;
  float* sX   = smem + OFF_X;
  float* sMsk = smem + OFF_MASK;

  const int tid  = threadIdx.x;
  const int wave = tid >> 5;
  const int lane = tid & 31;
  const int half = lane >> 4;    // 0: lanes 0-15, 1: lanes 16-31
  const int l16  = lane & 15;
  const int koff = half * 2;     // A/B K sub-offset per ISA 16x4 f32 layout

  const int b  = blockIdx.x >> 4;
  const int i0 = (blockIdx.x & 15) * ROWS;

  // ---- async staging: x1 tile (16x128) and mask tile (16x256) -> LDS,
  //      overlapped with the phase-0 WMMA below (which doesn't touch them) ----
  {
    const int row = tid >> 4;            // 0..15
    const int col = (tid & 15) * 8;      // 0..120
    const float* g = x1 + ((size_t)(b * N1D + i0 + row)) * FDIM + col;
    const unsigned off = (unsigned)((OFF_X1 + row * SP + col) * sizeof(float));
    stage_b128(smem, off,      g);
    stage_b128(smem, off + 16, g + 4);
  }
  {
    const int row = tid >> 4;
    const int col = (tid & 15) * 16;     // 0..240
    const float* g = ve + ((size_t)(b * N1D + i0 + row)) * N2D + col;
    const unsigned off = (unsigned)((OFF_MASK + row * SMP + col) * sizeof(float));
    stage_b128(smem, off,      g);
    stage_b128(smem, off + 16, g + 4);
    stage_b128(smem, off + 32, g + 8);
    stage_b128(smem, off + 48, g + 12);
  }

  // ---- phase 0: Mx2[b] = x2[b] @ Mw^T + Mb into LDS via WMMA (128 tiles / 8 waves) ----
  for (int t = wave; t < 128; t += 8) {
    const int m0 = (t >> 3) * 16;        // row tile in N2
    const int g0 = (t & 7) * 16;         // col tile in F
    const float* aB = x2 + ((size_t)(b * N2D + m0 + l16)) * FDIM + koff;  // A: x2 rows
    const float* bB = Mw + (size_t)(g0 + l16) * FDIM + koff;              // B[k][g] = Mw[g][k]
    v8f acc = {};
#pragma unroll 8
    for (int k = 0; k < FDIM; k += 4)
      acc = wmma_k4(*(const v2f*)(aB + k), *(const v2f*)(bB + k), acc);
    const float bias = Mb[g0 + l16];
#pragma unroll
    for (int vi = 0; vi < 8; ++vi) {
      const int row = m0 + vi + 8 * half;        // C layout: VGPR vi -> M = vi + 8*half
      sMx2[row * SP + g0 + l16] = acc[vi] + bias;
    }
  }
  wait_async0();        // staged tiles resident in LDS
  __syncthreads();

  // ---- phase 1: m2[i,f] = max_j mask[i,j]*Mx2[j,f]  (LDS-resident scan) ----
  // ~20% edge density: ballot lets the whole wave skip j-columns where both of
  // its rows are masked out (expected ~2.7x fewer inner iterations).
  {
    const int i  = tid >> 4;             // 0..15
    const int f0 = (tid & 15) * 8;       // 0..120
    float4 a0, a1;
    a0.x = a0.y = a0.z = a0.w = -3.4e38f; a1 = a0;
    const float* mrow = sMsk + i * SMP;
    bool sawZero = false;
    for (int j = 0; j < N2D; ++j) {
      const float m = mrow[j];           // broadcast within 16-thread group
      if (!wave_any_nonzero(m)) {        // wave-uniform: s_cbranch past dead column
        sawZero = true;
        continue;
      }
      const float4* vr = (const float4*)(sMx2 + j * SP + f0);
      const float4 v0 = vr[0];
      const float4 v1 = vr[1];
      a0.x = fmaxf(a0.x, m * v0.x);
      a0.y = fmaxf(a0.y, m * v0.y);
      a0.z = fmaxf(a0.z, m * v0.z);
      a0.w = fmaxf(a0.w, m * v0.w);
      a1.x = fmaxf(a1.x, m * v1.x);
      a1.y = fmaxf(a1.y, m * v1.y);
      a1.z = fmaxf(a1.z, m * v1.z);
      a1.w = fmaxf(a1.w, m * v1.w);
    }
    if (sawZero) {                       // skipped columns contribute 0-products
      a0.x = fmaxf(a0.x, 0.0f); a0.y = fmaxf(a0.y, 0.0f);
      a0.z = fmaxf(a0.z, 0.0f); a0.w = fmaxf(a0.w, 0.0f);
      a1.x = fmaxf(a1.x, 0.0f); a1.y = fmaxf(a1.y, 0.0f);
      a1.z = fmaxf(a1.z, 0.0f); a1.w = fmaxf(a1.w, 0.0f);
    }
    float4* xo = (float4*)(sX + i * SP + f0);
    xo[0] = a0; xo[1] = a1;
  }
  __syncthreads();

  // ---- phase 2: x = relu(x1 @ Ww^T + Wb + m2), in place in sX ----
  {
    const int g0 = wave * 16;                        // each wave owns one 16-col tile
    const float* aB = sX1 + l16 * SP + koff;         // A from LDS x1 tile
    const float* bB = Ww + (size_t)(g0 + l16) * FDIM + koff;
    v8f acc = {};
#pragma unroll 8
    for (int k = 0; k < FDIM; k += 4)
      acc = wmma_k4(*(const v2f*)(aB + k), *(const v2f*)(bB + k), acc);
    const float bias = Wb[g0 + l16];
#pragma unroll
    for (int vi = 0; vi < 8; ++vi) {
      const int row = vi + 8 * half;
      const int idx = row * SP + g0 + l16;           // unique owner per element
      sX[idx] = fmaxf(acc[vi] + bias + sX[idx], 0.0f);
    }
  }
  __syncthreads();

  // ---- phase 3: GRU gates via 6 WMMA chains per wave, then elementwise ----
  {
    const int f0 = wave * 16;
    const int fc = f0 + l16;
    const float* aXB = sX  + l16 * SP + koff;   // input x
    const float* aHB = sX1 + l16 * SP + koff;   // hidden x1
    const float* bI0 = wih + (size_t)fc * FDIM + koff;
    const float* bI1 = wih + (size_t)(fc + FDIM) * FDIM + koff;
    const float* bI2 = wih + (size_t)(fc + 2 * FDIM) * FDIM + koff;
    const float* bH0 = whh + (size_t)fc * FDIM + koff;
    const float* bH1 = whh + (size_t)(fc + FDIM) * FDIM + koff;
    const float* bH2 = whh + (size_t)(fc + 2 * FDIM) * FDIM + koff;
    __builtin_prefetch(bI0, 0, 1); __builtin_prefetch(bI1, 0, 1); __builtin_prefetch(bI2, 0, 1);
    __builtin_prefetch(bH0, 0, 1); __builtin_prefetch(bH1, 0, 1); __builtin_prefetch(bH2, 0, 1);

    v8f cI0 = {}, cI1 = {}, cI2 = {}, cH0 = {}, cH1 = {}, cH2 = {};
#pragma unroll 4
    for (int k = 0; k < FDIM; k += 4) {
      const v2f ax = *(const v2f*)(aXB + k);
      const v2f ah = *(const v2f*)(aHB + k);
      cI0 = wmma_k4(ax, *(const v2f*)(bI0 + k), cI0);
      cH0 = wmma_k4(ah, *(const v2f*)(bH0 + k), cH0);
      cI1 = wmma_k4(ax, *(const v2f*)(bI1 + k), cI1);
      cH1 = wmma_k4(ah, *(const v2f*)(bH1 + k), cH1);
      cI2 = wmma_k4(ax, *(const v2f*)(bI2 + k), cI2);
      cH2 = wmma_k4(ah, *(const v2f*)(bH2 + k), cH2);
    }
    const float bi0 = bih[fc], bi1 = bih[fc + FDIM], bi2 = bih[fc + 2 * FDIM];
    const float bh0 = bhh[fc], bh1 = bhh[fc + FDIM], bh2 = bhh[fc + 2 * FDIM];
#pragma unroll
    for (int vi = 0; vi < 8; ++vi) {
      const int row = vi + 8 * half;
      const float r  = sigm(cI0[vi] + bi0 + cH0[vi] + bh0);
      const float z  = sigm(cI1[vi] + bi1 + cH1[vi] + bh1);
      const float hn = cH2[vi] + bh2;
      const float n  = tanh_fast(cI2[vi] + bi2 + r * hn);
      const float hp = sX1[row * SP + fc];
      out[((size_t)(b * N1D + i0 + row)) * FDIM + fc] = (1.0f - z) * n + z * hp;
    }
  }
}

extern "C" void kernel_launch(void* const* d_in, const int* in_sizes, int n_in,
                              void* d_out, int out_size, void* d_ws, size_t ws_size,
                              hipStream_t stream) {
  (void)in_sizes; (void)n_in; (void)d_ws; (void)ws_size; (void)out_size;
  const float* x1  = (const float*)d_in[0];
  const float* x2  = (const float*)d_in[1];
  const float* ve  = (const float*)d_in[2];
  const float* Ww  = (const float*)d_in[3];
  const float* Wb  = (const float*)d_in[4];
  const float* Mw  = (const float*)d_in[5];
  const float* Mb  = (const float*)d_in[6];
  const float* wih = (const float*)d_in[7];
  const float* whh = (const float*)d_in[8];
  const float* bih = (const float*)d_in[9];
  const float* bhh = (const float*)d_in[10];
  float* out = (float*)d_out;

  const dim3 grid(8 * 16);   // (batch, 16-row i tile)
  const dim3 block(256);     // 8 wave32s
  const size_t shmem = (size_t)SMEM_FLOATS * sizeof(float);  // ~165 KB of the 320 KB WGP LDS
  inet_fused<<<grid, block, shmem, stream>>>(x1, x2, ve, Ww, Wb, Mw, Mb,
                                             wih, whh, bih, bhh, out);
}